// RandomEQ_13743895347207
// MI455X (gfx1250) — compile-verified
//
#include <hip/hip_runtime.h>
#include <hip/hip_bf16.h>

// ---------------------------------------------------------------------------
// RandomEQ: 3 cascaded biquads over clip[64][320000] (f32).
// Chunked-parallel IIR: per-chunk zero-state scan (Phase A), 6x6 state
// recurrence across chunks (Phase B), and a WMMA f32 16x16x4 correction
// matmul  Y += Phi(16x8) * S_init(8x16)  accumulated onto y_zero (Phase C).
// ---------------------------------------------------------------------------

typedef __attribute__((ext_vector_type(2))) float v2f;
typedef __attribute__((ext_vector_type(8))) float v8f;

constexpr int N_ROW = 64;
constexpr int T_LEN = 320000;
constexpr int K_CH  = 800;              // chunk length (multiple of 16)
constexpr int N_CH  = T_LEN / K_CH;     // 400 chunks per row (multiple of 16)

// workspace layout (in floats)
constexpr int WS_PHI   = 0;                     // [8][K_CH]  basis responses (rows 6,7 = 0)
constexpr int WS_M     = WS_PHI + 8 * K_CH;     // [6][6]     chunk transition matrix
constexpr int WS_SZERO = WS_M + 64;             // [64][400][8] zero-state chunk end states
constexpr int WS_SINIT = WS_SZERO + N_ROW * N_CH * 8; // [64][400][8] true chunk init states

struct EqCoefs { float b0, b1, b2, a1, a2; };

__device__ __forceinline__ void compute_coefs(const float* f, const float* g,
                                              const float* q, EqCoefs c[3]) {
  const float LN10 = 2.302585092994046f;
  const float PI   = 3.14159265358979323846f;
#pragma unroll
  for (int i = 0; i < 3; ++i) {
    float w0    = 2.f * PI * f[i] / 32000.f;
    float A     = expf(g[i] * (1.f / 40.f) * LN10);
    float alpha = sinf(w0) / (2.f * q[i]);
    float cs    = cosf(w0);
    float b0 = 1.f + alpha * A;
    float b1 = -2.f * cs;
    float b2 = 1.f - alpha * A;
    float a0 = 1.f + alpha / A;
    float a2 = 1.f - alpha / A;
    float inv = 1.f / a0;
    c[i] = {b0 * inv, b1 * inv, b2 * inv, b1 * inv, a2 * inv};
  }
}

// ---------------------------------------------------------------------------
// Basis kernel: 6 zero-input unit-state runs -> phi_j[n] (cascade output) and
// the 6x6 chunk-transition matrix M (final state of each run). Threads 6,7
// zero-fill the padded phi rows used by the second WMMA accumulation step.
// ---------------------------------------------------------------------------
__global__ void eq_basis_kernel(const float* __restrict__ freqs,
                                const float* __restrict__ gains,
                                const float* __restrict__ qs,
                                float* __restrict__ ws) {
  int j = threadIdx.x;
  float* PHI = ws + WS_PHI;
  if (j >= 6) {               // padded rows for the K=4..7 WMMA step
    for (int n = 0; n < K_CH; ++n) PHI[j * K_CH + n] = 0.f;
    return;
  }
  EqCoefs c[3];
  compute_coefs(freqs, gains, qs, c);

  float s0 = 0.f, s1 = 0.f, s2 = 0.f, s3 = 0.f, s4 = 0.f, s5 = 0.f;
  if (j == 0) s0 = 1.f; else if (j == 1) s1 = 1.f;
  else if (j == 2) s2 = 1.f; else if (j == 3) s3 = 1.f;
  else if (j == 4) s4 = 1.f; else s5 = 1.f;

  for (int n = 0; n < K_CH; ++n) {
    float y1 = -(c[0].a1 * s0 + c[0].a2 * s1);                       // x == 0
    float y2 = c[1].b0 * y1 + c[1].b1 * s0 + c[1].b2 * s1
             - c[1].a1 * s2 - c[1].a2 * s3;
    float y3 = c[2].b0 * y2 + c[2].b1 * s2 + c[2].b2 * s3
             - c[2].a1 * s4 - c[2].a2 * s5;
    PHI[j * K_CH + n] = y3;
    s1 = s0; s0 = y1;
    s3 = s2; s2 = y2;
    s5 = s4; s4 = y3;
  }
  float* M = ws + WS_M;       // M[i][j] = final state component i of run j
  M[0 * 6 + j] = s0; M[1 * 6 + j] = s1; M[2 * 6 + j] = s2;
  M[3 * 6 + j] = s3; M[4 * 6 + j] = s4; M[5 * 6 + j] = s5;
}

// ---------------------------------------------------------------------------
// Phase A: one thread per (row, chunk). Zero-y-state cascade scan with real
// inputs (x-history read directly from the input buffer). Writes y_zero to
// d_out and the chunk end-state to ws.
// ---------------------------------------------------------------------------
__global__ void eq_phaseA_kernel(const float* __restrict__ in,
                                 const float* __restrict__ freqs,
                                 const float* __restrict__ gains,
                                 const float* __restrict__ qs,
                                 float* __restrict__ out,
                                 float* __restrict__ ws) {
  int tid = blockIdx.x * blockDim.x + threadIdx.x;   // exact grid: 64*400
  int row = tid / N_CH;
  int ch  = tid - row * N_CH;
  size_t base = (size_t)row * T_LEN + (size_t)ch * K_CH;

  EqCoefs c[3];
  compute_coefs(freqs, gains, qs, c);

  float xm1 = (ch > 0) ? in[base - 1] : 0.f;
  float xm2 = (ch > 0) ? in[base - 2] : 0.f;
  float y1m1 = 0.f, y1m2 = 0.f, y2m1 = 0.f, y2m2 = 0.f, y3m1 = 0.f, y3m2 = 0.f;

  for (int n = 0; n < K_CH; ++n) {
    float x  = in[base + n];
    float y1 = c[0].b0 * x + c[0].b1 * xm1 + c[0].b2 * xm2
             - c[0].a1 * y1m1 - c[0].a2 * y1m2;
    float y2 = c[1].b0 * y1 + c[1].b1 * y1m1 + c[1].b2 * y1m2
             - c[1].a1 * y2m1 - c[1].a2 * y2m2;
    float y3 = c[2].b0 * y2 + c[2].b1 * y2m1 + c[2].b2 * y2m2
             - c[2].a1 * y3m1 - c[2].a2 * y3m2;
    out[base + n] = y3;
    xm2 = xm1; xm1 = x;
    y1m2 = y1m1; y1m1 = y1;
    y2m2 = y2m1; y2m1 = y2;
    y3m2 = y3m1; y3m1 = y3;
  }
  float* sz = ws + WS_SZERO + (size_t)tid * 8;
  sz[0] = y1m1; sz[1] = y1m2; sz[2] = y2m1; sz[3] = y2m2; sz[4] = y3m1; sz[5] = y3m2;
}

// ---------------------------------------------------------------------------
// Phase B: one thread per row; sequential over 400 chunks:
//   S_init[c] = S ; S = S_zero_end[c] + M * S
// Pads S_init components 6,7 with zeros (consumed by the second WMMA step).
// ---------------------------------------------------------------------------
__global__ void eq_phaseB_kernel(float* __restrict__ ws) {
  int r = threadIdx.x;                      // 64 threads
  const float* M = ws + WS_M;
  float m[36];
#pragma unroll
  for (int i = 0; i < 36; ++i) m[i] = M[i];

  float S[6] = {0.f, 0.f, 0.f, 0.f, 0.f, 0.f};
  const float* SZ = ws + WS_SZERO;
  float*       SI = ws + WS_SINIT;
  for (int ch = 0; ch < N_CH; ++ch) {
    size_t b = ((size_t)r * N_CH + ch) * 8;
#pragma unroll
    for (int i = 0; i < 6; ++i) SI[b + i] = S[i];
    SI[b + 6] = 0.f; SI[b + 7] = 0.f;
    float Sn[6];
#pragma unroll
    for (int i = 0; i < 6; ++i) {
      float acc = SZ[b + i];
#pragma unroll
      for (int jj = 0; jj < 6; ++jj) acc += m[i * 6 + jj] * S[jj];
      Sn[i] = acc;
    }
#pragma unroll
    for (int i = 0; i < 6; ++i) S[i] = Sn[i];
  }
}

// ---------------------------------------------------------------------------
// Phase C: WMMA correction. One wave per 16(time) x 16(chunk) tile of one row:
//   D = Phi(16x4) * S(4x16) + [Phi(16x4) * S(4x16)] + y_zero   (two K=4 steps)
// A layout (16x4 f32): v0 = K0 (lanes 0-15) / K2 (lanes 16-31); v1 = K1/K3.
// B laid out symmetrically; C/D: row M = r + 8*half, col N = lane&15.
// ---------------------------------------------------------------------------
__global__ void eq_phaseC_kernel(float* __restrict__ out,
                                 const float* __restrict__ ws) {
  int tile = blockIdx.x * 8 + (threadIdx.x >> 5);  // 8 waves / block, exact grid
  int lane = threadIdx.x & 31;
  int half = lane >> 4;
  int l16  = lane & 15;

  // tiles per row: (N_CH/16) * (K_CH/16) = 25 * 50 = 1250
  int row = tile / 1250;
  int rem = tile - row * 1250;
  int ct  = rem / 50;       // chunk-tile
  int tt  = rem - ct * 50;  // time-tile
  int c0  = ct * 16;
  int t0  = tt * 16;

  const float* PHI = ws + WS_PHI;
  const float* SI  = ws + WS_SINIT;

  v2f a0, a1, b0, b1;
  a0.x = PHI[(half ? 2 : 0) * K_CH + t0 + l16];
  a0.y = PHI[(half ? 3 : 1) * K_CH + t0 + l16];
  a1.x = PHI[(half ? 6 : 4) * K_CH + t0 + l16];
  a1.y = PHI[(half ? 7 : 5) * K_CH + t0 + l16];

  const float* sp = SI + ((size_t)row * N_CH + c0 + l16) * 8;
  b0.x = sp[half ? 2 : 0];
  b0.y = sp[half ? 3 : 1];
  b1.x = sp[half ? 6 : 4];
  b1.y = sp[half ? 7 : 5];

  // each lane's 8 C/D elements are contiguous: times t0 + 8*half + [0..7]
  float* po = out + (size_t)row * T_LEN + (size_t)(c0 + l16) * K_CH
                  + t0 + 8 * half;
  v8f c;
#pragma unroll
  for (int r = 0; r < 8; ++r) c[r] = po[r];

  c = __builtin_amdgcn_wmma_f32_16x16x4_f32(false, a0, false, b0,
                                            (short)0, c, false, false);
  c = __builtin_amdgcn_wmma_f32_16x16x4_f32(false, a1, false, b1,
                                            (short)0, c, false, false);

#pragma unroll
  for (int r = 0; r < 8; ++r) po[r] = c[r];
}

// ---------------------------------------------------------------------------
extern "C" void kernel_launch(void* const* d_in, const int* in_sizes, int n_in,
                              void* d_out, int out_size, void* d_ws, size_t ws_size,
                              hipStream_t stream) {
  const float* clip  = (const float*)d_in[0];
  const float* freqs = (const float*)d_in[1];
  const float* gains = (const float*)d_in[2];
  const float* qs    = (const float*)d_in[3];
  float* out = (float*)d_out;
  float* ws  = (float*)d_ws;   // needs ~1.7 MB

  eq_basis_kernel<<<1, 8, 0, stream>>>(freqs, gains, qs, ws);
  // 64 rows * 400 chunks = 25600 threads
  eq_phaseA_kernel<<<100, 256, 0, stream>>>(clip, freqs, gains, qs, out, ws);
  eq_phaseB_kernel<<<1, 64, 0, stream>>>(ws);
  // 64 * 1250 tiles = 80000 waves = 10000 blocks of 8 waves
  eq_phaseC_kernel<<<10000, 256, 0, stream>>>(out, ws);
}